// MambaBlock_22333829939828
// MI455X (gfx1250) — compile-verified
//
#include <hip/hip_runtime.h>
#include <hip/hip_bf16.h>

// Mamba block, MI455X (gfx1250), fp32 end-to-end via V_WMMA_F32_16X16X4_F32.
#define BATCHN  4
#define SEQLEN  2048
#define DMODEL  1024
#define DSTATE  16
#define DCONV   4
#define DINNER  2048
#define MROWS   (BATCHN * SEQLEN)   // 8192 flattened (b, l) rows

typedef __attribute__((ext_vector_type(2))) float v2f;
typedef __attribute__((ext_vector_type(8))) float v8f;

// ---------------------------------------------------------------------------
// GEMM: C[M,N] = A[M,K] * W[N,K]^T  (row-major), fp32 WMMA 16x16x4.
// One wave computes a 64x64 tile: 4x4 v8f accumulators, 16 WMMAs per K=4 step.
// K-loop is software-pipelined: the 8 fragment loads for step k+4 are issued
// as a clause before the 16 WMMAs of step k, so load latency overlaps the
// matrix pipe instead of serializing with it.
// EPI==1: C = softplus(C + bias[n]) with inline v_exp/v_log (no libcalls).
// Requires M % 64 == 0 (true here) and K % 4 == 0; N may be ragged (N=16).
// ---------------------------------------------------------------------------
template <int EPI>
__global__ __launch_bounds__(256, 1) void gemm_nt_f32_wmma(
    const float* __restrict__ A, const float* __restrict__ W,
    float* __restrict__ C, int M, int N, int K,
    const float* __restrict__ bias)
{
  const int lane = threadIdx.x & 31;
  const int wid  = blockIdx.x * 8 + (threadIdx.x >> 5);
  const int Mt = M >> 6;
  const int Nt = (N + 63) >> 6;
  if (wid >= Mt * Nt) return;            // whole wave exits; EXEC stays all-ones
  const int tn = wid / Mt;
  const int tm = wid - tn * Mt;
  const int m0 = tm << 6;
  const int n0 = tn << 6;

  const int half = lane >> 4;            // 0: K={0,1}  1: K={2,3} within step
  const int lq   = lane & 15;            // row (A) / col (B) within 16-tile
  const int kk   = half * 2;

  // 32-bit element offsets (largest buffer = 33.5M floats -> fits easily)
  unsigned aoff[4], boff[4];
#pragma unroll
  for (int t = 0; t < 4; ++t) {
    aoff[t] = (unsigned)(m0 + t * 16 + lq) * (unsigned)K + kk;
    int col = n0 + t * 16 + lq; if (col > N - 1) col = N - 1;  // clamp (N=16)
    boff[t] = (unsigned)col * (unsigned)K + kk;
  }

  v8f acc[4][4];
  const v8f vzero = {};
#pragma unroll
  for (int i = 0; i < 4; ++i)
#pragma unroll
    for (int j = 0; j < 4; ++j) acc[i][j] = vzero;

  // Prologue: fragments for k = 0.
  v2f a[4], b[4];
#pragma unroll
  for (int t = 0; t < 4; ++t) { a[t] = *(const v2f*)(A + aoff[t]); aoff[t] += 4; }
#pragma unroll
  for (int t = 0; t < 4; ++t) { b[t] = *(const v2f*)(W + boff[t]); boff[t] += 4; }

  // Pipelined main loop: prefetch k+4 while multiplying k.
#pragma unroll 1
  for (int k = 0; k < K - 4; k += 4) {
    v2f an[4], bn[4];
#pragma unroll
    for (int t = 0; t < 4; ++t) { an[t] = *(const v2f*)(A + aoff[t]); aoff[t] += 4; }
#pragma unroll
    for (int t = 0; t < 4; ++t) { bn[t] = *(const v2f*)(W + boff[t]); boff[t] += 4; }
#pragma unroll
    for (int mt = 0; mt < 4; ++mt)
#pragma unroll
      for (int nt = 0; nt < 4; ++nt)
        acc[mt][nt] = __builtin_amdgcn_wmma_f32_16x16x4_f32(
            false, a[mt], false, b[nt], (short)0, acc[mt][nt], false, false);
#pragma unroll
    for (int t = 0; t < 4; ++t) { a[t] = an[t]; b[t] = bn[t]; }
  }

  // Epilogue step (k = K-4).
#pragma unroll
  for (int mt = 0; mt < 4; ++mt)
#pragma unroll
    for (int nt = 0; nt < 4; ++nt)
      acc[mt][nt] = __builtin_amdgcn_wmma_f32_16x16x4_f32(
          false, a[mt], false, b[nt], (short)0, acc[mt][nt], false, false);

  // Bias per n-subtile (lane-dependent, invariant over mt/r) hoisted once.
  float bv[4];
  if (EPI == 1) {
#pragma unroll
    for (int nt = 0; nt < 4; ++nt) {
      int n = n0 + nt * 16 + lq;
      bv[nt] = (n < N) ? bias[n] : 0.0f;
    }
  }

  // C/D layout: elem r of v8f -> M = r + 8*half, N = lq (within 16x16 tile).
  unsigned crow = (unsigned)(m0 + half * 8) * (unsigned)N + (unsigned)(n0 + lq);
#pragma unroll
  for (int mt = 0; mt < 4; ++mt) {
#pragma unroll
    for (int r = 0; r < 8; ++r) {
      unsigned row = crow + (unsigned)(mt * 16 + r) * (unsigned)N;
#pragma unroll
      for (int nt = 0; nt < 4; ++nt) {
        int n = n0 + nt * 16 + lq;
        if (n < N) {
          float v = acc[mt][nt][r];
          if (EPI == 1) {   // softplus, branch-free inline intrinsics
            v += bv[nt];
            v = fmaxf(v, 0.0f) + __logf(1.0f + __expf(-fabsf(v)));
          }
          C[row + nt * 16] = v;
        }
      }
    }
  }
}

// ---------------------------------------------------------------------------
// Causal depthwise conv1d (k=4, left-pad 3) + bias + SiLU.
// Reads x_ssm = xr[:, :, 0:DINNER] (row stride 2*DINNER), writes xc[b,l,c].
// ---------------------------------------------------------------------------
__global__ __launch_bounds__(256) void conv_silu_kernel(
    const float* __restrict__ xr, const float* __restrict__ Wc,
    const float* __restrict__ bc, float* __restrict__ xc)
{
  int idx = blockIdx.x * 256 + threadIdx.x;   // = (b*SEQ + s)*DINNER + c
  int c   = idx & (DINNER - 1);
  int bs  = idx >> 11;                        // b*SEQ + s
  int s   = bs & (SEQLEN - 1);
  float acc = bc[c];
#pragma unroll
  for (int j = 0; j < DCONV; ++j) {
    int sj = s - (DCONV - 1) + j;
    if (sj >= 0)
      acc += xr[(unsigned)(bs - (DCONV - 1) + j) * (2u * DINNER) + (unsigned)c]
             * Wc[c * DCONV + j];
  }
  float sig = 1.0f / (1.0f + __expf(-acc));
  xc[idx] = acc * sig;
}

// ---------------------------------------------------------------------------
// Selective scan fused with the output gate  y = (sum_h + D*x) * silu(res).
// One thread per (batch, channel, state): 131072 threads = 4096 wave32s.
// 16 states of a channel occupy 16 adjacent lanes; y_t is a 4-step shfl_xor
// butterfly over that group (ds_swizzle on gfx1250).
// ---------------------------------------------------------------------------
__global__ __launch_bounds__(256) void scan_kernel(
    const float* __restrict__ xc, const float* __restrict__ dt,
    const float* __restrict__ Bm, const float* __restrict__ A_log,
    const float* __restrict__ Dv, const float* __restrict__ xr,
    float* __restrict__ y)
{
  int t = blockIdx.x * 256 + threadIdx.x;     // = ((b*DINNER + c)*16 + s)
  int s = t & 15;
  int c = (t >> 4) & (DINNER - 1);
  int b = t >> 15;

  float Acs = -__expf(A_log[c * DSTATE + s]); // A = -exp(A_log)
  float Dc  = Dv[c];

  unsigned m0 = (unsigned)b * SEQLEN;
  unsigned ox  = m0 * DINNER + (unsigned)c;        // into xc / dt / y
  unsigned oB  = m0 * DSTATE + (unsigned)s;        // into Bm
  unsigned orr = m0 * (2u * DINNER) + DINNER + (unsigned)c;  // res half of xr

  float h = 0.0f;
#pragma unroll 1
  for (int l = 0; l < SEQLEN; ++l) {
    float xt  = xc[ox];
    float dtt = dt[ox];
    float Bv  = Bm[oB];
    float dA  = __expf(dtt * Acs);
    h = dA * h + dtt * Bv * xt;

    float sum = h;
#pragma unroll
    for (int off = 1; off < DSTATE; off <<= 1)
      sum += __shfl_xor(sum, off, 32);

    if (s == 0) {
      float res = xr[orr];
      float g   = res / (1.0f + __expf(-res));   // silu(res)
      y[ox] = (sum + Dc * xt) * g;
    }
    ox  += DINNER;
    oB  += DSTATE;
    orr += 2 * DINNER;
  }
}

// ---------------------------------------------------------------------------
extern "C" void kernel_launch(void* const* d_in, const int* in_sizes, int n_in,
                              void* d_out, int out_size, void* d_ws, size_t ws_size,
                              hipStream_t stream)
{
  (void)in_sizes; (void)n_in; (void)out_size; (void)ws_size;
  const float* x      = (const float*)d_in[0];
  const float* W_in   = (const float*)d_in[1];
  const float* W_conv = (const float*)d_in[2];
  const float* b_conv = (const float*)d_in[3];
  const float* W_x    = (const float*)d_in[4];
  const float* W_dt   = (const float*)d_in[5];
  const float* b_dt   = (const float*)d_in[6];
  const float* W_out  = (const float*)d_in[7];
  const float* A_log  = (const float*)d_in[8];
  const float* Dv     = (const float*)d_in[9];
  float* out = (float*)d_out;

  // Workspace layout (floats): xr[8192*4096] xc[8192*2048] dt[8192*2048]
  //                            Bm[8192*16]   y[8192*2048]   (~336 MB)
  float* ws = (float*)d_ws;
  float* xr = ws;
  float* xc = xr + (size_t)MROWS * 2 * DINNER;
  float* dt = xc + (size_t)MROWS * DINNER;
  float* Bm = dt + (size_t)MROWS * DINNER;
  float* y  = Bm + (size_t)MROWS * DSTATE;

  auto gemmGrid = [](int M, int N) {
    int tiles = (M >> 6) * ((N + 63) >> 6);
    return (tiles + 7) / 8;                    // 8 waves / 256-thread block
  };

  // 1) xr = x @ W_in^T                      [8192 x 4096]
  gemm_nt_f32_wmma<0><<<gemmGrid(MROWS, 2 * DINNER), 256, 0, stream>>>(
      x, W_in, xr, MROWS, 2 * DINNER, DMODEL, nullptr);

  // 2) xc = silu(causal_dwconv(x_ssm) + b)  [8192 x 2048]
  conv_silu_kernel<<<(MROWS * DINNER) / 256, 256, 0, stream>>>(
      xr, W_conv, b_conv, xc);

  // 3) dt = softplus(xc @ W_dt^T + b_dt)    [8192 x 2048]
  gemm_nt_f32_wmma<1><<<gemmGrid(MROWS, DINNER), 256, 0, stream>>>(
      xc, W_dt, dt, MROWS, DINNER, DINNER, b_dt);

  // 4) Bm = xc @ W_x[16:32]^T  (delta half of W_x is dead code)  [8192 x 16]
  gemm_nt_f32_wmma<0><<<gemmGrid(MROWS, DSTATE), 256, 0, stream>>>(
      xc, W_x + (size_t)DSTATE * DINNER, Bm, MROWS, DSTATE, DINNER, nullptr);

  // 5) y = gate(selective_scan(...))        [8192 x 2048]
  scan_kernel<<<(BATCHN * DINNER * DSTATE) / 256, 256, 0, stream>>>(
      xc, dt, Bm, A_log, Dv, xr, y);

  // 6) out = y @ W_out^T                    [8192 x 1024]
  gemm_nt_f32_wmma<0><<<gemmGrid(MROWS, DMODEL), 256, 0, stream>>>(
      y, W_out, out, MROWS, DMODEL, DINNER, nullptr);
}